// My_RelGraphConv_87926570484531
// MI455X (gfx1250) — compile-verified
//
#include <hip/hip_runtime.h>
#include <hip/hip_bf16.h>

typedef __attribute__((ext_vector_type(2))) float v2f;
typedef __attribute__((ext_vector_type(8))) float v8f;

#define N_NODES   10000
#define N_EDGES   160000
#define FEAT      256
#define NUM_RELS  16
#define WG_PER_REL 16   // workgroups per relation for the edge GEMM

// gfx1250 async global->LDS copy path (ASYNCcnt-tracked), with fallback
#if defined(__HIP_DEVICE_COMPILE__) && __has_builtin(__builtin_amdgcn_global_load_async_to_lds_b32)
#define ASYNC_LDS 1
#else
#define ASYNC_LDS 0
#endif

typedef __attribute__((address_space(1))) int gas_i32;
typedef __attribute__((address_space(3))) int las_i32;

__device__ __forceinline__ void wait_async_copies() {
#if ASYNC_LDS
#if __has_builtin(__builtin_amdgcn_s_wait_asynccnt)
    __builtin_amdgcn_s_wait_asynccnt(0);
#else
    asm volatile("s_wait_asynccnt 0x0" ::: "memory");
#endif
#endif
}

// ---------------------------------------------------------------------------
// Kernel 1: zero degree arrays, relation histogram, cursors
// ---------------------------------------------------------------------------
__global__ __launch_bounds__(256) void k_init(float* degD, float* degS, int* off, int* cur) {
    int i = blockIdx.x * 256 + threadIdx.x;
    if (i < N_NODES) { degD[i] = 0.f; degS[i] = 0.f; }
    if (i < NUM_RELS + 1) off[i] = 0;
    if (i < NUM_RELS)     cur[i] = 0;
}

// ---------------------------------------------------------------------------
// Kernel 2: degrees (float atomics, exact below 2^24) + per-relation counts
// ---------------------------------------------------------------------------
__global__ __launch_bounds__(256) void k_deg(const int* __restrict__ src,
                                             const int* __restrict__ dst,
                                             const int* __restrict__ et,
                                             float* degD, float* degS, int* off) {
    int e = blockIdx.x * 256 + threadIdx.x;
    if (e < N_EDGES) {
        atomicAdd(&degD[dst[e]], 1.f);
        atomicAdd(&degS[src[e]], 1.f);
        atomicAdd(&off[et[e] + 1], 1);
    }
}

// ---------------------------------------------------------------------------
// Kernel 3: invert degrees (0 if deg==0) + tiny 16-way exclusive scan
// ---------------------------------------------------------------------------
__global__ __launch_bounds__(256) void k_finalize(float* degD, float* degS, int* off, int* cur) {
    for (int i = threadIdx.x; i < N_NODES; i += 256) {
        float d = degD[i]; degD[i] = d > 0.f ? 1.f / d : 0.f;
        float s = degS[i]; degS[i] = s > 0.f ? 1.f / s : 0.f;
    }
    if (threadIdx.x == 0) {
        // off[r+1] currently holds count of relation r; off[0] == 0
        for (int r = 1; r <= NUM_RELS; ++r) off[r] += off[r - 1];
        for (int r = 0; r < NUM_RELS; ++r)  cur[r] = off[r];
    }
}

// ---------------------------------------------------------------------------
// Kernel 4: counting-sort scatter -> perm[] groups edge ids by relation
// ---------------------------------------------------------------------------
__global__ __launch_bounds__(256) void k_scatter(const int* __restrict__ et, int* cur, int* perm) {
    int e = blockIdx.x * 256 + threadIdx.x;
    if (e < N_EDGES) {
        int p = atomicAdd(&cur[et[e]], 1);
        perm[p] = e;
    }
}

// ---------------------------------------------------------------------------
// B-panel fill, pre-swizzled into v_wmma_f32_16x16x4_f32 B-fragment order:
//   buf[ks][nt][lane] : lane<16 -> (W[k0+0,n], W[k0+1,n])
//                       lane>=16 -> (W[k0+2,n], W[k0+3,n]),  n = nt*16 + lane%16
// 32 consecutive float2 per fragment -> 64 dwords -> all 64 LDS banks.
// Async path: two per-lane B32 ASYNCcnt-tracked copies per float2 (no VGPR trip).
// ---------------------------------------------------------------------------
__device__ __forceinline__ void fill_panel(float2 (*buf)[16][32],
                                           const float* __restrict__ Wr,
                                           int krow0, int tid) {
#pragma unroll
    for (int j = 0; j < 8; ++j) {
        int idx = j * 256 + tid;     // 0..2047 float2 slots
        int ls  = idx & 31;
        int nt  = (idx >> 5) & 15;
        int ks  = idx >> 9;
        int n   = (nt << 4) + (ls & 15);
        int kr  = krow0 + (ks << 2) + ((ls >> 4) << 1);
        const float* g = Wr + kr * FEAT + n;
        float*       l = (float*)&buf[ks][nt][ls];
#if ASYNC_LDS
        __builtin_amdgcn_global_load_async_to_lds_b32(
            (gas_i32*)g, (las_i32*)l, 0, 0);
        __builtin_amdgcn_global_load_async_to_lds_b32(
            (gas_i32*)(g + FEAT), (las_i32*)(l + 1), 0, 0);
#else
        float2 bv;
        bv.x = g[0];
        bv.y = g[FEAT];
        buf[ks][nt][ls] = bv;
#endif
    }
}

// ---------------------------------------------------------------------------
// Unified WMMA GEMM:
//   mode 0 (self loop):  out[row]  = 2 * (feat[row] @ W) + bias        (store)
//   mode 1 (edge pass):  out[tgt] += (feat[gat] @ W[rel] + bias)*norm  (atomic)
// 8 wave32s / block; each wave computes a 16-row x 256-col tile with
// V_WMMA_F32_16X16X4_F32. B double-buffered in LDS; panel k+1 fill overlaps
// panel k compute (single barrier per panel).
// ---------------------------------------------------------------------------
__global__ __launch_bounds__(256) void k_rgcn_gemm(
    const float* __restrict__ feat, const float* __restrict__ W,
    const float* __restrict__ bias, float* __restrict__ out,
    const int* __restrict__ perm, const int* __restrict__ gidx,
    const int* __restrict__ sidx, const float* __restrict__ invdeg,
    const int* __restrict__ off, int mode)
{
    __shared__ float2 Bs[2][4][16][32];   // 2 x 16 KB double buffer

    const int lane = threadIdx.x & 31;
    const int w    = threadIdx.x >> 5;

    int r, chunk0, cstride, base, cnt;
    if (mode) {
        r       = blockIdx.x & (NUM_RELS - 1);
        chunk0  = blockIdx.x >> 4;
        cstride = gridDim.x  >> 4;
        base    = off[r];
        cnt     = off[r + 1] - base;
    } else {
        r = 0; chunk0 = blockIdx.x; cstride = gridDim.x; base = 0; cnt = N_NODES;
    }
    const float* Wr = W + (size_t)r * (FEAT * FEAT);

    const int nchunks = (cnt + 127) >> 7;   // 128 rows (edges) per block-chunk
    for (int chunk = chunk0; chunk < nchunks; chunk += cstride) {
        // ---- per-lane row metadata (lane l handles logical row l&15) ----
        int  row   = (chunk << 7) + (w << 4) + (lane & 15);
        bool valid = row < cnt;
        int gnode, tgt; float nrm, bml;
        if (mode) {
            int p = base + (valid ? row : 0);
            int e = perm[p];
            gnode = gidx[e];
            tgt   = sidx[e];
            nrm   = valid ? invdeg[tgt] : 0.f;   // (proj + bias) * norm
            bml   = nrm;
        } else {
            gnode = valid ? row : 0;
            tgt   = row;
            nrm   = 2.f;    // s = s + s + bias  ->  2*s + bias
            bml   = 1.f;
        }
        int vflag = valid ? 1 : 0;
        // A-fragment base: lanes 0-15 read K=k0,k0+1; lanes 16-31 read K=k0+2,k0+3
        const float* arow = feat + (size_t)gnode * FEAT + ((lane >> 4) << 1);

        v8f acc[16];
        const v8f zero = {0.f,0.f,0.f,0.f,0.f,0.f,0.f,0.f};
        #pragma unroll
        for (int nt = 0; nt < 16; ++nt) acc[nt] = zero;

        // prime first panel of the double buffer
        fill_panel(Bs[0], Wr, 0, threadIdx.x);

        for (int kp = 0; kp < 16; ++kp) {          // 16 K-panels of 16
            const int cb = kp & 1;
            wait_async_copies();                    // own async B copies done
            __syncthreads();                        // panel visible; prev reads done
            if (kp < 15)                            // overlap next fill with compute
                fill_panel(Bs[cb ^ 1], Wr, (kp + 1) << 4, threadIdx.x);

            const int krow0 = kp << 4;
            #pragma unroll
            for (int ks = 0; ks < 4; ++ks) {
                const float2 av = *(const float2*)(arow + krow0 + (ks << 2));
                v2f a; a.x = av.x; a.y = av.y;
                #pragma unroll
                for (int nt = 0; nt < 16; ++nt) {
                    float2 bv = Bs[cb][ks][nt][lane];   // one ds_load_b64, bank-free
                    v2f b; b.x = bv.x; b.y = bv.y;
                    acc[nt] = __builtin_amdgcn_wmma_f32_16x16x4_f32(
                        /*neg_a=*/false, a, /*neg_b=*/false, b,
                        /*c_mod=*/(short)0, acc[nt],
                        /*reuse_a=*/false, /*reuse_b=*/false);
                }
            }
        }

        // ---- epilogue: C layout => VGPR v: lanes0-15 M=v, lanes16-31 M=v+8 ----
        #pragma unroll 4
        for (int nt = 0; nt < 16; ++nt) {
            int   col = (nt << 4) + (lane & 15);
            float bv  = bias[col];
            #pragma unroll
            for (int v = 0; v < 8; ++v) {
                int   rowi = v + ((lane >> 4) << 3);       // 0..15
                float nr = __shfl(nrm,   rowi, 32);
                float bm = __shfl(bml,   rowi, 32);
                int   tg = __shfl(tgt,   rowi, 32);
                int   vd = __shfl(vflag, rowi, 32);
                float val = acc[nt][v] * nr + bv * bm;
                if (vd) {
                    float* dp = out + (size_t)tg * FEAT + col;
                    if (mode) atomicAdd(dp, val);   // global_atomic_add_f32 (L2-resident out)
                    else      *dp = val;
                }
            }
        }
    }
}

// ---------------------------------------------------------------------------
extern "C" void kernel_launch(void* const* d_in, const int* in_sizes, int n_in,
                              void* d_out, int out_size, void* d_ws, size_t ws_size,
                              hipStream_t stream) {
    const float* feat  = (const float*)d_in[0];
    const float* Wf    = (const float*)d_in[1];
    const float* Wb    = (const float*)d_in[2];
    const float* loopW = (const float*)d_in[3];
    const float* fbias = (const float*)d_in[4];
    const float* bbias = (const float*)d_in[5];
    const float* sbias = (const float*)d_in[6];
    const int*   src   = (const int*)d_in[7];
    const int*   dst   = (const int*)d_in[8];
    const int*   et    = (const int*)d_in[9];
    float*       out   = (float*)d_out;

    char* ws = (char*)d_ws;
    float* degD = (float*)ws; ws += N_NODES * sizeof(float);   // 1/in-deg by dst
    float* degS = (float*)ws; ws += N_NODES * sizeof(float);   // 1/in-deg by src
    int*   off  = (int*)ws;   ws += 32 * sizeof(int);          // relation offsets [17]
    int*   cur  = (int*)ws;   ws += 32 * sizeof(int);          // scatter cursors [16]
    int*   perm = (int*)ws;                                    // edge ids by relation [E]

    k_init    <<<(N_NODES + 255) / 256, 256, 0, stream>>>(degD, degS, off, cur);
    k_deg     <<<(N_EDGES + 255) / 256, 256, 0, stream>>>(src, dst, et, degD, degS, off);
    k_finalize<<<1, 256, 0, stream>>>(degD, degS, off, cur);
    k_scatter <<<(N_EDGES + 255) / 256, 256, 0, stream>>>(et, cur, perm);

    // self loop first (plain stores initialize out), then edge passes atomic-add
    k_rgcn_gemm<<<(N_NODES + 127) / 128, 256, 0, stream>>>(
        feat, loopW, sbias, out, nullptr, nullptr, nullptr, nullptr, nullptr, 0);
    // forward: gather feat[src], weights Wf[etype], norm 1/indeg_dst(dst), scatter dst
    k_rgcn_gemm<<<NUM_RELS * WG_PER_REL, 256, 0, stream>>>(
        feat, Wf, fbias, out, perm, src, dst, degD, off, 1);
    // backward: gather feat[dst], weights Wb[etype], norm 1/indeg_src(src), scatter src
    k_rgcn_gemm<<<NUM_RELS * WG_PER_REL, 256, 0, stream>>>(
        feat, Wb, bbias, out, perm, dst, src, degS, off, 1);
}